// ONNX_ORT_MASK_36240934043985
// MI455X (gfx1250) — compile-verified
//
#include <hip/hip_runtime.h>
#include <hip/hip_bf16.h>
#include <stdint.h>

#define NUM_ANCHORS 25200
#define NUM_CLASSES 80
#define MAX_OBJ     100
#define ATTN_RES    14
#define MASK_RES    56
#define NUM_BASE    5
#define BASE_RES    160
#define X0_STRIDE   85                                  // 4 box + 1 conf + 80 cls
#define ATTN_LEN    (NUM_BASE * ATTN_RES * ATTN_RES)    // 980 floats
#define OUT_STRIDE  (1 + 4 + 1 + 1 + MASK_RES * MASK_RES) // 3143

typedef float v2f  __attribute__((ext_vector_type(2)));
typedef float v8f  __attribute__((ext_vector_type(8)));
typedef unsigned int u32x4 __attribute__((ext_vector_type(4)));
typedef int   i32x4 __attribute__((ext_vector_type(4)));
typedef int   i32x8 __attribute__((ext_vector_type(8)));

// ---------------------------------------------------------------------------
// Kernel 1: gathered box conversion (WMMA f32 16x16x4) + conf*cls max/argmax.
// One block, 256 threads (8 waves). Waves 0..6 each own one 16-row WMMA tile
// (7*16 = 112 >= 100 rows); rows >= 100 are clamped duplicates and discarded
// at store time so EXEC stays all-ones for the WMMA itself.
//
// B (4x16) = CONVERT zero-padded; built arithmetically (no memory, no
// branches):  column n, K-pair {0,1}: magnitude 1.0; K-pair {2,3}: magnitude
// -0.5 (n<2) / +0.5 (n>=2); parity of n selects which K slot is nonzero.
// ---------------------------------------------------------------------------
__global__ __launch_bounds__(256) void boxes_scores_kernel(
    const float* __restrict__ x0, const int* __restrict__ sel_idx,
    float* __restrict__ out, float* __restrict__ ws_boxes)
{
    const int tid  = threadIdx.x;
    const int wave = tid >> 5;
    const int lane = tid & 31;

    if (wave < 7) {                      // wave-uniform branch: EXEC all ones
        const int half = lane >> 4;      // 0 -> K {0,1}, 1 -> K {2,3}
        const int m    = lane & 15;
        const int row  = wave * 16 + m;
        const int r    = row < MAX_OBJ ? row : MAX_OBJ - 1;
        const int idx  = sel_idx[r];
        const float* px = x0 + (long)idx * X0_STRIDE;

        // A: 16x4 f32, lane layout per ISA 7.12.2 (VGPR0: K=0|2, VGPR1: K=1|3)
        const int ko = half << 1;        // 0 or 2
        v2f a;
        a.x = px[ko];
        a.y = px[ko + 1];

        // B: branch-free construction of CONVERT padded to 4x16
        const int  col   = m;
        const bool inmat = (col < 4);
        const bool odd   = (col & 1) != 0;
        const float mag  = half ? ((col < 2) ? -0.5f : 0.5f) : 1.0f;
        v2f b;
        b.x = (inmat && !odd) ? mag : 0.0f;  // K slots 0 / 2
        b.y = (inmat &&  odd) ? mag : 0.0f;  // K slots 1 / 3
        v8f c = {0.f, 0.f, 0.f, 0.f, 0.f, 0.f, 0.f, 0.f};
        v8f d = __builtin_amdgcn_wmma_f32_16x16x4_f32(
            /*neg_a=*/false, a, /*neg_b=*/false, b,
            /*c_mod=*/(short)0, c, /*reuse_a=*/false, /*reuse_b=*/false);

        // D layout: VGPR v -> M = v (+8 for lanes 16..31), N = lane&15
        const int obase = wave * 16 + (half ? 8 : 0);
        if (inmat) {
            #pragma unroll
            for (int v = 0; v < 8; ++v) {
                const int orow = obase + v;
                if (orow < MAX_OBJ) {
                    const float val = d[v];
                    ws_boxes[orow * 4 + col] = val;
                    out[(long)orow * OUT_STRIDE + 1 + col] = val;
                }
            }
        }
    }

    // conf * cls, max + first-argmax over 80 classes (one thread per ROI)
    if (tid < MAX_OBJ) {
        const int idx = sel_idx[tid];
        const float* px = x0 + (long)idx * X0_STRIDE;
        const float conf = px[4];
        float best = px[5] * conf;
        int bestc = 0;
        #pragma unroll 4
        for (int j = 1; j < NUM_CLASSES; ++j) {
            const float s = px[5 + j] * conf;
            if (s > best) { best = s; bestc = j; }
        }
        float* o = out + (long)tid * OUT_STRIDE;
        o[0] = 0.0f;            // batch index (sel_batch is all zero)
        o[5] = (float)bestc;    // category
        o[6] = best;            // score
    }
}

// ---------------------------------------------------------------------------
// Kernel 2: one block per ROI. TDM DMAs the 980-float attention row into LDS
// (tensor_load_to_lds + s_wait_tensorcnt), bases rows are prefetched into L2,
// then each thread produces 56x56/256 mask pixels: ROI-align (2x2 samples,
// bilinear, edge clamp) + attn 14->56 bilinear upsample + softmax(5) + dot +
// sigmoid.
// ---------------------------------------------------------------------------
__global__ __launch_bounds__(256) void mask_kernel(
    const float* __restrict__ attn, const float* __restrict__ bases,
    const int* __restrict__ sel_idx, const float* __restrict__ ws_boxes,
    float* __restrict__ out)
{
    __shared__ float s_attn[ATTN_LEN];   // 5 x 14 x 14, 3920 B

    const int r   = blockIdx.x;
    const int idx = sel_idx[r];

    // ---- Tensor Data Mover: contiguous 1 x 980 f32 tile, global -> LDS ----
    if (threadIdx.x < 32) {              // one issuing wave
        const unsigned long long gaddr =
            (unsigned long long)(uintptr_t)(attn + (long)idx * ATTN_LEN);
        const unsigned int laddr =
            (unsigned int)(uintptr_t)(void*)s_attn;  // low 32 bits = LDS offset

        u32x4 g0;
        g0[0] = 1u;                                       // count=1, user D#
        g0[1] = laddr;                                    // lds_addr
        g0[2] = (unsigned int)(gaddr & 0xFFFFFFFFull);    // global_addr[31:0]
        g0[3] = (unsigned int)((gaddr >> 32) & 0x1FFFFFFull)
              | 0x80000000u;                              // addr[56:32] | type=2

        i32x8 g1;
        g1[0] = 0x00020000;                               // data_size = 4 bytes
        g1[1] = (int)((ATTN_LEN & 0xFFFF) << 16);         // tensor_dim0 lo
        g1[2] = (int)(((ATTN_LEN >> 16) & 0xFFFF) | (1u << 16)); // dim0 hi | tensor_dim1=1
        g1[3] = (int)((unsigned)ATTN_LEN << 16);          // tile_dim0 = 980
        g1[4] = 1;                                        // tile_dim1 = 1
        g1[5] = ATTN_LEN;                                 // tensor_dim0_stride
        g1[6] = 0;
        g1[7] = 0;

        i32x4 gz  = {0, 0, 0, 0};                         // groups 2/3 unused (2D)
        i32x8 gz8 = {0, 0, 0, 0, 0, 0, 0, 0};             // trailing group (clang-23 form)
        __builtin_amdgcn_tensor_load_to_lds(g0, g1, gz, gz, gz8, 0);
        __builtin_amdgcn_s_wait_tensorcnt(0);
    }
    __syncthreads();

    // Per-block-uniform ROI geometry (POOLER_SCALE = 0.25)
    const float bx1   = ws_boxes[r * 4 + 0] * 0.25f;
    const float by1   = ws_boxes[r * 4 + 1] * 0.25f;
    const float bx2   = ws_boxes[r * 4 + 2] * 0.25f;
    const float by2   = ws_boxes[r * 4 + 3] * 0.25f;
    const float roi_w = fmaxf(bx2 - bx1, 1.0f);
    const float roi_h = fmaxf(by2 - by1, 1.0f);

    // Warm L2 with the ROI's bases rows (global_prefetch_b8)
    {
        int ry0 = max(0, (int)by1);
        int ry1 = min(BASE_RES - 1, (int)(by1 + roi_h) + 1);
        for (int c = 0; c < NUM_BASE; ++c) {
            const float* bc = bases + c * (BASE_RES * BASE_RES);
            for (int row = ry0 + (int)threadIdx.x; row <= ry1; row += blockDim.x)
                __builtin_prefetch(bc + row * BASE_RES, 0, 0);
        }
    }

    for (int p = threadIdx.x; p < MASK_RES * MASK_RES; p += blockDim.x) {
        const int oy = p / MASK_RES;
        const int ox = p - oy * MASK_RES;

        // ---- ROI align: 2x2 samples, bilinear with edge clamp ----
        float pooled[NUM_BASE] = {0.f, 0.f, 0.f, 0.f, 0.f};
        #pragma unroll
        for (int sy = 0; sy < 2; ++sy) {
            const float ty = ((float)(2 * oy + sy) + 0.5f) * (1.0f / 112.0f);
            float y = by1 + ty * roi_h;
            y = fminf(fmaxf(y, 0.0f), 159.0f);
            const float y0f = floorf(y);
            const int   y0  = (int)y0f;
            const int   y1i = min(y0 + 1, BASE_RES - 1);
            const float wy1 = y - y0f, wy0 = 1.0f - wy1;
            #pragma unroll
            for (int sx = 0; sx < 2; ++sx) {
                const float tx = ((float)(2 * ox + sx) + 0.5f) * (1.0f / 112.0f);
                float x = bx1 + tx * roi_w;
                x = fminf(fmaxf(x, 0.0f), 159.0f);
                const float x0f = floorf(x);
                const int   x0i = (int)x0f;
                const int   x1i = min(x0i + 1, BASE_RES - 1);
                const float wx1 = x - x0f, wx0 = 1.0f - wx1;
                const float w00 = wy0 * wx0, w01 = wy0 * wx1;
                const float w10 = wy1 * wx0, w11 = wy1 * wx1;
                const int o00 = y0  * BASE_RES + x0i, o01 = y0  * BASE_RES + x1i;
                const int o10 = y1i * BASE_RES + x0i, o11 = y1i * BASE_RES + x1i;
                #pragma unroll
                for (int c = 0; c < NUM_BASE; ++c) {
                    const float* bc = bases + c * (BASE_RES * BASE_RES);
                    pooled[c] = fmaf(w00, bc[o00],
                                fmaf(w01, bc[o01],
                                fmaf(w10, bc[o10],
                                fmaf(w11, bc[o11], pooled[c]))));
                }
            }
        }

        // ---- attn 14 -> 56 bilinear (half-pixel centers), from LDS ----
        float ay = ((float)oy + 0.5f) * 0.25f - 0.5f;
        float ax = ((float)ox + 0.5f) * 0.25f - 0.5f;
        ay = fminf(fmaxf(ay, 0.0f), 13.0f);
        ax = fminf(fmaxf(ax, 0.0f), 13.0f);
        const float ayf = floorf(ay); const int ay0 = (int)ayf;
        const float axf = floorf(ax); const int ax0 = (int)axf;
        const int ay1 = min(ay0 + 1, ATTN_RES - 1);
        const int ax1 = min(ax0 + 1, ATTN_RES - 1);
        const float vy1 = ay - ayf, vy0 = 1.0f - vy1;
        const float vx1 = ax - axf, vx0 = 1.0f - vx1;

        float a[NUM_BASE];
        float amax = -1e30f;
        #pragma unroll
        for (int c = 0; c < NUM_BASE; ++c) {
            const float* ac = s_attn + c * (ATTN_RES * ATTN_RES);
            const float v = vy0 * (vx0 * ac[ay0 * ATTN_RES + ax0] +
                                   vx1 * ac[ay0 * ATTN_RES + ax1]) +
                            vy1 * (vx0 * ac[ay1 * ATTN_RES + ax0] +
                                   vx1 * ac[ay1 * ATTN_RES + ax1]);
            a[c] = v;
            amax = fmaxf(amax, v);
        }
        float asum = 0.0f;
        #pragma unroll
        for (int c = 0; c < NUM_BASE; ++c) { a[c] = __expf(a[c] - amax); asum += a[c]; }
        float acc = 0.0f;
        #pragma unroll
        for (int c = 0; c < NUM_BASE; ++c) acc = fmaf(pooled[c], a[c], acc);
        acc = acc * 0.25f / asum;                 // 0.25 = mean over 2x2 samples

        out[(long)r * OUT_STRIDE + 7 + p] = 1.0f / (1.0f + __expf(-acc));
    }
}

extern "C" void kernel_launch(void* const* d_in, const int* in_sizes, int n_in,
                              void* d_out, int out_size, void* d_ws, size_t ws_size,
                              hipStream_t stream) {
    const float* x0      = (const float*)d_in[0];   // (1, 25200, 85)
    const float* attn    = (const float*)d_in[1];   // (1, 25200, 980)
    const float* bases   = (const float*)d_in[2];   // (1, 5, 160, 160)
    // d_in[3] = sel_batch (all zeros, batch==1 -> unused)
    const int*   sel_idx = (const int*)d_in[4];     // (100,)

    float* out      = (float*)d_out;                // 100 x 3143
    float* ws_boxes = (float*)d_ws;                 // 100 x 4 scratch

    boxes_scores_kernel<<<1, 256, 0, stream>>>(x0, sel_idx, out, ws_boxes);
    mask_kernel<<<MAX_OBJ, 256, 0, stream>>>(attn, bases, sel_idx, ws_boxes, out);
}